// PAElla_86011015070176
// MI455X (gfx1250) — compile-verified
//
#include <hip/hip_runtime.h>
#include <stdint.h>

#define B_SZ 256
#define T_SZ 8192
#define E_CH 8
#define KW   15
#define N2   4096                      // packed complex FFT length (T/2)
#define TPI  6.28318530717958647692f
#define PI_F 3.14159265358979323846f

#define USE_ASYNC_LDS 1                // gfx1250 async global->LDS copies

typedef __attribute__((ext_vector_type(2))) float v2f;
typedef __attribute__((ext_vector_type(8))) float v8f;

static __device__ __forceinline__ int imin(int a, int b) { return a < b ? a : b; }
static __device__ __forceinline__ int imax(int a, int b) { return a > b ? a : b; }

static __device__ __forceinline__ void async_wait0() {
#if __has_builtin(__builtin_amdgcn_s_wait_asynccnt)
  __builtin_amdgcn_s_wait_asynccnt(0);
#else
  asm volatile("s_wait_asynccnt 0" ::: "memory");
#endif
}

// per-lane async copy of one dword global -> LDS (gfx1250 GLOBAL_LOAD_ASYNC_TO_LDS_B32)
static __device__ __forceinline__ void async_g2l_b32(const float* gsrc, float* ldst) {
  unsigned lds_off = (unsigned)(uintptr_t)ldst;           // generic LDS addr: low 32 bits = LDS offset
  unsigned long long ga = (unsigned long long)(uintptr_t)gsrc;
  asm volatile("global_load_async_to_lds_b32 %0, %1, off" :: "v"(lds_off), "v"(ga) : "memory");
}

// ---------------------------------------------------------------------------
// K1: per-channel dilated conv stack  x -> h (LDS only) -> latent
//     dilation d = 2^e, SAME zero padding for both convs.
// ---------------------------------------------------------------------------
__global__ __launch_bounds__(256) void conv_latent_kernel(
    const float* __restrict__ x, const float* __restrict__ W1, const float* __restrict__ b1,
    const float* __restrict__ W2, const float* __restrict__ b2, float* __restrict__ latent)
{
  const int TILE = 2048;
  int be = blockIdx.y;
  int b = be >> 3, e = be & 7;
  int d = 1 << e;
  int tile0 = blockIdx.x * TILE;
  int tid = threadIdx.x;
  int H1 = 7 * d, H2 = 14 * d;

  __shared__ float sx[TILE + 2 * 1792];          // x tile + halo (max e=7)
  __shared__ float sh[2][TILE + 2 * 896];        // hidden 2-channel tile + halo

  float w1[2][KW], w2[2][KW];
  #pragma unroll
  for (int c = 0; c < 2; ++c)
    #pragma unroll
    for (int k = 0; k < KW; ++k) {
      w1[c][k] = W1[(e * 2 + c) * KW + k];       // (E,2,1,K)
      w2[c][k] = W2[(e * 2 + c) * KW + k];       // (E,1,2,K) flattens identically
    }
  float bb1_0 = b1[e * 2 + 0], bb1_1 = b1[e * 2 + 1], bb2 = b2[e];

  const float* xrow = x + (size_t)b * T_SZ;
  int nx = TILE + 2 * H2;
  int x0 = tile0 - H2;

  // prefetch hint for the next tile (gfx1250 global_prefetch_b8)
  if (tid < 64) {
    int g = imin(tile0 + TILE + tid * 32, T_SZ - 1);
    __builtin_prefetch(xrow + g, 0, 0);
  }

#if USE_ASYNC_LDS
  for (int i = tid; i < nx; i += 256) {
    int g = x0 + i;
    int gc = imin(imax(g, 0), T_SZ - 1);         // clamp; OOB entries fixed below
    async_g2l_b32(xrow + gc, &sx[i]);
  }
  async_wait0();
  __syncthreads();
  for (int i = tid; i < nx; i += 256) {
    int g = x0 + i;
    if (g < 0 || g >= T_SZ) sx[i] = 0.0f;        // SAME zero padding
  }
  __syncthreads();
#else
  for (int i = tid; i < nx; i += 256) {
    int g = x0 + i;
    sx[i] = (g >= 0 && g < T_SZ) ? xrow[g] : 0.0f;
  }
  __syncthreads();
#endif

  // conv1: 1 -> 2 channels (dilated).  h[c][j] at global pos h0+j
  int nh = TILE + 2 * H1;
  int h0 = tile0 - H1;
  for (int j = tid; j < nh; j += 256) {
    int t = h0 + j;
    float a0 = 0.0f, a1 = 0.0f;
    if (t >= 0 && t < T_SZ) {                    // h is zero-padded outside [0,T)
      a0 = bb1_0; a1 = bb1_1;
      #pragma unroll
      for (int k = 0; k < KW; ++k) {
        float xv = sx[j + k * d];
        a0 = fmaf(w1[0][k], xv, a0);
        a1 = fmaf(w1[1][k], xv, a1);
      }
    }
    sh[0][j] = a0; sh[1][j] = a1;
  }
  __syncthreads();

  // conv2: 2 -> 1 channel (dilated)
  float* lrow = latent + (size_t)be * T_SZ;
  for (int i = tid; i < TILE; i += 256) {
    float acc = bb2;
    #pragma unroll
    for (int k = 0; k < KW; ++k) {
      acc = fmaf(w2[0][k], sh[0][i + k * d], acc);
      acc = fmaf(w2[1][k], sh[1][i + k * d], acc);
    }
    lrow[tile0 + i] = acc;
  }
}

// ---------------------------------------------------------------------------
// K2: rfft power-spectrum stats per (b,e) via packed real FFT (4096-pt complex,
//     in-place radix-2 DIT in LDS) + even/odd unpack of bins 1..4096.
// ---------------------------------------------------------------------------
__global__ __launch_bounds__(256) void fft_stats_kernel(
    const float* __restrict__ latent, float* __restrict__ fOut,
    float* __restrict__ aOut, float* __restrict__ bOut)
{
  __shared__ float2 z[N2];          // 32 KB
  __shared__ float2 tw[N2 / 2];     // 16 KB twiddles e^{-2pi i k/N2}
  __shared__ float2 rsum[256];
  int be = blockIdx.x, tid = threadIdx.x;
  const float* row = latent + (size_t)be * T_SZ;

  for (int i = tid; i < N2; i += 256)
    z[i] = ((const float2*)row)[i];             // z_n = x_{2n} + i x_{2n+1}
  for (int i = tid; i < N2 / 2; i += 256) {
    float s, c;
    sincosf(-TPI * (float)i / (float)N2, &s, &c);
    tw[i] = make_float2(c, s);
  }
  __syncthreads();

  // bit-reverse permutation (12 bits)
  for (int i = tid; i < N2; i += 256) {
    int j = (int)(__brev((unsigned)i) >> 20);
    if (j > i) { float2 t = z[i]; z[i] = z[j]; z[j] = t; }
  }
  __syncthreads();

  for (int len = 2; len <= N2; len <<= 1) {
    int half = len >> 1;
    int tstep = N2 / len;
    for (int q = tid; q < N2 / 2; q += 256) {
      int grp = q / half;
      int pos = q - grp * half;
      int i0 = grp * len + pos, i1 = i0 + half;
      float2 w = tw[pos * tstep];
      float2 u = z[i0], v = z[i1];
      float2 vw = make_float2(v.x * w.x - v.y * w.y, v.x * w.y + v.y * w.x);
      z[i0] = make_float2(u.x + vw.x, u.y + vw.y);
      z[i1] = make_float2(u.x - vw.x, u.y - vw.y);
    }
    __syncthreads();
  }

  // unpack rfft bins k = 1..4096; freqs[k] = k/WINDOW = k/2
  float psum = 0.0f, fsum = 0.0f;
  for (int k = tid + 1; k <= N2; k += 256) {
    float2 Zk = (k < N2) ? z[k] : z[0];
    float2 Zr = z[(N2 - k) & (N2 - 1)];
    float Ex = 0.5f * (Zk.x + Zr.x);
    float Ey = 0.5f * (Zk.y - Zr.y);
    float Ox = 0.5f * (Zk.y + Zr.y);
    float Oy = 0.5f * (Zr.x - Zk.x);
    float s, c;
    sincosf(-PI_F * (float)k / (float)N2, &s, &c);  // e^{-2pi i k/T}
    float Xr = Ex + c * Ox - s * Oy;
    float Xi = Ey + c * Oy + s * Ox;
    float pw = Xr * Xr + Xi * Xi;
    psum += pw;
    fsum = fmaf(0.5f * (float)k, pw, fsum);
  }
  rsum[tid] = make_float2(psum, fsum);
  __syncthreads();
  for (int s = 128; s > 0; s >>= 1) {
    if (tid < s) {
      rsum[tid].x += rsum[tid + s].x;
      rsum[tid].y += rsum[tid + s].y;
    }
    __syncthreads();
  }
  if (tid == 0) {
    float ps = rsum[0].x, fs = rsum[0].y;
    fOut[be] = fs / ps;
    aOut[be] = 2.0f * sqrtf(ps) / (float)T_SZ;
    bOut[be] = (z[0].x + z[0].y) / (float)T_SZ;   // X_0.real = sum(x)
  }
}

// ---------------------------------------------------------------------------
// K3: per-channel Linear(T,2) as V_WMMA_F32_16X16X4_F32, then phase = atan2/2pi.
//     block = 8 waves; each wave owns a K slice of 1024; LDS cross-wave reduce.
// ---------------------------------------------------------------------------
__global__ __launch_bounds__(256) void phase_wmma_kernel(
    const float* __restrict__ latent, const float* __restrict__ fcW,
    const float* __restrict__ fcb, float* __restrict__ pOut)
{
  __shared__ float cacc[8 * 256];
  int e = blockIdx.x & 7;
  int mtile = blockIdx.x >> 3;
  int tid = threadIdx.x;
  int wave = tid >> 5, lane = tid & 31;
  int hi = lane >> 4, n = lane & 15;             // A: M = lane&15 ; B/C/D: N = lane&15
  int brow = mtile * 16 + n;

  const float* Arow = latent + ((size_t)brow * E_CH + e) * T_SZ;   // latent[b][e][*]
  float bmask = (n < 2) ? 1.0f : 0.0f;                             // pad N: 2 -> 16
  const float* Brow = fcW + ((size_t)e * 2 + (n < 2 ? n : 0)) * T_SZ;

  v8f acc = {0.f, 0.f, 0.f, 0.f, 0.f, 0.f, 0.f, 0.f};
  int k0 = wave * 1024, kend = k0 + 1024;
  #pragma unroll 4
  for (int k = k0; k < kend; k += 4) {
    int ka = k + 2 * hi;                         // A/B VGPR0,1 hold K = ka, ka+1
    float2 av = *(const float2*)(Arow + ka);
    float2 bv = *(const float2*)(Brow + ka);
    v2f A;  A.x = av.x;          A.y = av.y;
    v2f Bv; Bv.x = bv.x * bmask; Bv.y = bv.y * bmask;
    acc = __builtin_amdgcn_wmma_f32_16x16x4_f32(
        /*neg_a=*/false, A, /*neg_b=*/false, Bv,
        /*c_mod=*/(short)0, acc, /*reuse_a=*/false, /*reuse_b=*/false);
  }

  // spill each wave's 16x16 C tile (VGPR j, half hi -> M = j + 8*hi, N = n)
  #pragma unroll
  for (int j = 0; j < 8; ++j) {
    int m = j + 8 * hi;
    cacc[wave * 256 + m * 16 + n] = acc[j];
  }
  __syncthreads();
  float tot = 0.0f;
  #pragma unroll
  for (int w = 0; w < 8; ++w) tot += cacc[w * 256 + tid];
  cacc[tid] = tot;                               // element tid = (m = tid/16, n = tid%16)
  __syncthreads();
  if (tid < 16) {
    float v0 = cacc[tid * 16 + 0] + fcb[e * 2 + 0];
    float v1 = cacc[tid * 16 + 1] + fcb[e * 2 + 1];
    pOut[(size_t)(mtile * 16 + tid) * E_CH + e] = atan2f(v1, v0) * (1.0f / TPI);
  }
}

// ---------------------------------------------------------------------------
// K4: sinusoid synthesis  sig = a*sin(2pi*(f*args + p)) + b_off
// ---------------------------------------------------------------------------
__global__ __launch_bounds__(256) void sig_kernel(
    const float* __restrict__ fv, const float* __restrict__ av,
    const float* __restrict__ pv, const float* __restrict__ bv,
    float* __restrict__ sig)
{
  size_t idx = ((size_t)blockIdx.x * 256 + threadIdx.x) * 4;
  int t = (int)(idx & (T_SZ - 1));
  int be = (int)(idx >> 13);
  float f = fv[be], a = av[be], p = pv[be], b0 = bv[be];
  const float step = 2.0f / (float)(T_SZ - 1);   // linspace(-1, 1, T)
  float4 r;
  #pragma unroll
  for (int m = 0; m < 4; ++m) {
    float arg = fmaf((float)(t + m), step, -1.0f);
    ((float*)&r)[m] = fmaf(a, sinf(TPI * fmaf(f, arg, p)), b0);
  }
  *(float4*)(sig + idx) = r;
}

// ---------------------------------------------------------------------------
// K5: fused 3-level grouped deconv tree 8ch -> 4 -> 2 -> 1, all in LDS.
// ---------------------------------------------------------------------------
__global__ __launch_bounds__(256) void deconv_tree_kernel(
    const float* __restrict__ sig,
    const float* __restrict__ dW0, const float* __restrict__ db0,
    const float* __restrict__ dW1, const float* __restrict__ db1,
    const float* __restrict__ dW2, const float* __restrict__ db2,
    float* __restrict__ out)
{
  const int TILE = 1024;
  const int SS = TILE + 42, S1 = TILE + 28, S2 = TILE + 14;
  __shared__ float ssig[8 * (1024 + 42)];
  __shared__ float sy1[4 * (1024 + 28)];
  __shared__ float sy2[2 * (1024 + 14)];
  int b = blockIdx.y;
  int tile0 = blockIdx.x * TILE;
  int tid = threadIdx.x;
  const float* srow = sig + (size_t)b * E_CH * T_SZ;

  for (int idx = tid; idx < 8 * SS; idx += 256) {
    int c = idx / SS, j = idx - c * SS;
    int t = tile0 - 21 + j;
    ssig[idx] = (t >= 0 && t < T_SZ) ? srow[(size_t)c * T_SZ + t] : 0.0f;
  }
  __syncthreads();

  for (int idx = tid; idx < 4 * S1; idx += 256) {       // level 0: groups=4
    int g = idx / S1, j = idx - g * S1;
    int t = tile0 - 14 + j;
    float acc = 0.0f;
    if (t >= 0 && t < T_SZ) {
      acc = db0[g];
      #pragma unroll
      for (int c = 0; c < 2; ++c) {
        const float* wp = dW0 + (g * 2 + c) * KW;
        const float* sp = ssig + (2 * g + c) * SS + j;
        #pragma unroll
        for (int k = 0; k < KW; ++k) acc = fmaf(wp[k], sp[k], acc);
      }
    }
    sy1[idx] = acc;
  }
  __syncthreads();

  for (int idx = tid; idx < 2 * S2; idx += 256) {       // level 1: groups=2
    int g = idx / S2, j = idx - g * S2;
    int t = tile0 - 7 + j;
    float acc = 0.0f;
    if (t >= 0 && t < T_SZ) {
      acc = db1[g];
      #pragma unroll
      for (int c = 0; c < 2; ++c) {
        const float* wp = dW1 + (g * 2 + c) * KW;
        const float* sp = sy1 + (2 * g + c) * S1 + j;
        #pragma unroll
        for (int k = 0; k < KW; ++k) acc = fmaf(wp[k], sp[k], acc);
      }
    }
    sy2[idx] = acc;
  }
  __syncthreads();

  for (int i = tid; i < TILE; i += 256) {               // level 2: groups=1
    float acc = db2[0];
    #pragma unroll
    for (int c = 0; c < 2; ++c) {
      const float* wp = dW2 + c * KW;
      const float* sp = sy2 + c * S2 + i;
      #pragma unroll
      for (int k = 0; k < KW; ++k) acc = fmaf(wp[k], sp[k], acc);
    }
    out[(size_t)b * T_SZ + tile0 + i] = acc;
  }
}

// ---------------------------------------------------------------------------
extern "C" void kernel_launch(void* const* d_in, const int* in_sizes, int n_in,
                              void* d_out, int out_size, void* d_ws, size_t ws_size,
                              hipStream_t stream)
{
  (void)in_sizes; (void)n_in; (void)out_size; (void)d_ws; (void)ws_size;
  const float* x   = (const float*)d_in[0];
  const float* W1  = (const float*)d_in[1];
  const float* b1  = (const float*)d_in[2];
  const float* W2  = (const float*)d_in[3];
  const float* b2  = (const float*)d_in[4];
  const float* fcW = (const float*)d_in[5];
  const float* fcb = (const float*)d_in[6];
  const float* dW0 = (const float*)d_in[7];
  const float* db0 = (const float*)d_in[8];
  const float* dW1 = (const float*)d_in[9];
  const float* db1 = (const float*)d_in[10];
  const float* dW2 = (const float*)d_in[11];
  const float* db2 = (const float*)d_in[12];

  float* out    = (float*)d_out;                                  // (B,T)
  float* latent = out + (size_t)B_SZ * T_SZ;                      // (B,E,T)
  float* sig    = latent + (size_t)B_SZ * E_CH * T_SZ;            // (B,E,T)
  float* pOut   = sig + (size_t)B_SZ * E_CH * T_SZ;               // (B,E,1)
  float* fOut   = pOut + B_SZ * E_CH;
  float* aOut   = fOut + B_SZ * E_CH;
  float* bOut   = aOut + B_SZ * E_CH;

  conv_latent_kernel<<<dim3(T_SZ / 2048, B_SZ * E_CH), 256, 0, stream>>>(x, W1, b1, W2, b2, latent);
  fft_stats_kernel<<<B_SZ * E_CH, 256, 0, stream>>>(latent, fOut, aOut, bOut);
  phase_wmma_kernel<<<E_CH * (B_SZ / 16), 256, 0, stream>>>(latent, fcW, fcb, pOut);
  sig_kernel<<<(B_SZ * E_CH * T_SZ) / (256 * 4), 256, 0, stream>>>(fOut, aOut, pOut, bOut, sig);
  deconv_tree_kernel<<<dim3(T_SZ / 1024, B_SZ), 256, 0, stream>>>(sig, dW0, db0, dW1, db1, dW2, db2, out);
}